// RecommendationLoss_86165633892908
// MI455X (gfx1250) — compile-verified
//
#include <hip/hip_runtime.h>
#include <math.h>

typedef float v2f __attribute__((ext_vector_type(2)));
typedef float v8f __attribute__((ext_vector_type(8)));

#define B_ 4096
#define T_ 8
#define M_ 10
#define D_ 128
#define E_ 256
#define NROWS (B_ * T_)        /* 32768 (b,t) rows / tiles */
#define NELEM (B_ * T_ * M_)   /* 327680 */
#define NPAIRS 45              /* M*(M-1)/2 */
#define EPS 1e-8f

#define K1_BLOCKS 128
#define K2_BLOCKS 1024
#define K3_BLOCKS 4096         /* 8 waves per block -> 32768 tiles */

/* workspace layout in floats */
#define WS1_OFF 0                          /* K1_BLOCKS * 4 : {sq,bce,phys,rank} */
#define WS2_OFF (K1_BLOCKS * 4)            /* K2_BLOCKS    : cos sums            */
#define WS3_OFF (WS2_OFF + K2_BLOCKS)      /* NROWS        : per-tile div sums   */

// ---------------------------------------------------------------------------
// Kernel 1: MSE + BCE + phys + pairwise margin-rank over [B,T,M] arrays.
// One thread per (b,t) row of M=10. Deterministic LDS tree -> 4 partials/block.
// ---------------------------------------------------------------------------
__global__ void k1_scores(const float* __restrict__ rs, const float* __restrict__ ts,
                          const float* __restrict__ fs, const float* __restrict__ tfs,
                          const float* __restrict__ rk, const float* __restrict__ trk,
                          float* __restrict__ ws) {
    const int row = blockIdx.x * blockDim.x + threadIdx.x;
    float sq = 0.f, bce = 0.f, phys = 0.f, rank = 0.f;
    if (row < NROWS) {
        const int base = row * M_;
        float r[M_], t[M_];
#pragma unroll
        for (int m = 0; m < M_; ++m) {
            const float a = rs[base + m], b = ts[base + m];
            const float d = a - b;
            sq += d * d;
            const float f = fs[base + m];
            const float p = fminf(fmaxf(f, 1e-7f), 1.0f - 1e-7f);
            const float y = tfs[base + m];
            bce -= y * logf(p) + (1.0f - y) * log1pf(-p);
            phys += fmaxf(0.7f - f, 0.0f);
            r[m] = rk[base + m];
            t[m] = trk[base + m];
        }
#pragma unroll
        for (int i = 0; i < M_; ++i)
#pragma unroll
            for (int j = i + 1; j < M_; ++j) {
                const float dtr = t[i] - t[j];
                const float pref = (dtr > 0.f) ? 1.f : ((dtr < 0.f) ? -1.f : 0.f);
                rank += fmaxf(0.1f - pref * (r[i] - r[j]), 0.0f);
            }
    }
    __shared__ float s0[256], s1[256], s2[256], s3[256];
    const int tid = threadIdx.x;
    s0[tid] = sq; s1[tid] = bce; s2[tid] = phys; s3[tid] = rank;
    __syncthreads();
    for (int off = 128; off > 0; off >>= 1) {
        if (tid < off) {
            s0[tid] += s0[tid + off]; s1[tid] += s1[tid + off];
            s2[tid] += s2[tid + off]; s3[tid] += s3[tid + off];
        }
        __syncthreads();
    }
    if (tid == 0) {
        float* o = ws + WS1_OFF + blockIdx.x * 4;
        o[0] = s0[0]; o[1] = s1[0]; o[2] = s2[0]; o[3] = s3[0];
    }
}

// ---------------------------------------------------------------------------
// Kernel 2: explanation cosine over E=256 rows (dominant 672 MB of traffic).
// One wave32 per row, two coalesced float4 sweeps per operand.
// ---------------------------------------------------------------------------
__global__ void k2_expl(const float* __restrict__ ja, const float* __restrict__ te,
                        float* __restrict__ ws) {
    const int lane = threadIdx.x & 31;
    const int wave = threadIdx.x >> 5;
    const int gw = blockIdx.x * 8 + wave;
    const int nw = gridDim.x * 8;
    float csum = 0.f;
    for (int row = gw; row < NELEM; row += nw) {
        const float* a = ja + (size_t)row * E_;
        const float* b = te + (size_t)row * E_;
        const float4 a0 = *(const float4*)(a + 4 * lane);
        const float4 a1 = *(const float4*)(a + 128 + 4 * lane);
        const float4 b0 = *(const float4*)(b + 4 * lane);
        const float4 b1 = *(const float4*)(b + 128 + 4 * lane);
        float dot = a0.x * b0.x + a0.y * b0.y + a0.z * b0.z + a0.w * b0.w
                  + a1.x * b1.x + a1.y * b1.y + a1.z * b1.z + a1.w * b1.w;
        float na = a0.x * a0.x + a0.y * a0.y + a0.z * a0.z + a0.w * a0.w
                 + a1.x * a1.x + a1.y * a1.y + a1.z * a1.z + a1.w * a1.w;
        float nb = b0.x * b0.x + b0.y * b0.y + b0.z * b0.z + b0.w * b0.w
                 + b1.x * b1.x + b1.y * b1.y + b1.z * b1.z + b1.w * b1.w;
        for (int off = 16; off > 0; off >>= 1) {
            dot += __shfl_xor(dot, off, 32);
            na  += __shfl_xor(na,  off, 32);
            nb  += __shfl_xor(nb,  off, 32);
        }
        if (lane == 0)
            csum += dot / (fmaxf(sqrtf(na), EPS) * fmaxf(sqrtf(nb), EPS));
    }
    __shared__ float ssum[8];
    if (lane == 0) ssum[wave] = csum;
    __syncthreads();
    if (threadIdx.x == 0) {
        float s = 0.f;
        for (int w = 0; w < 8; ++w) s += ssum[w];
        ws[WS2_OFF + blockIdx.x] = s;
    }
}

// ---------------------------------------------------------------------------
// Kernel 3: diversity via V_WMMA_F32_16X16X4_F32. One wave per (b,t) tile:
// G = N*N^T with N = 10x128 tile zero-padded to 16 rows; 32 chained WMMAs
// over K=128. A-layout (lane=M, vgpr=K, half=K+2) mirrored for B means the
// same v2f registers serve as both A and B(=N^T). Normalization folded in
// post-hoc via diag rsqrt; strict-upper-triangle masked sum.
// ---------------------------------------------------------------------------
__global__ void k3_div(const float* __restrict__ rec, float* __restrict__ ws) {
    const int lane = threadIdx.x & 31;
    const int wave = threadIdx.x >> 5;
    const int tile = blockIdx.x * 8 + wave;       /* 0 .. 32767 */
    const int m16  = lane & 15;
    const int hi   = lane >> 4;
    const int mcl  = (m16 < M_) ? m16 : (M_ - 1);
    const bool valid = (m16 < M_);
    const float* rowp = rec + (size_t)tile * (M_ * D_) + (size_t)mcl * D_ + 2 * hi;

    v8f acc = {0.f, 0.f, 0.f, 0.f, 0.f, 0.f, 0.f, 0.f};
#pragma unroll
    for (int s = 0; s < D_ / 4; ++s) {
        const float2 v = *(const float2*)(rowp + 4 * s);
        v2f a;
        a.x = valid ? v.x : 0.f;
        a.y = valid ? v.y : 0.f;
        /* D = A*B + C ; symmetric operands: same registers for A and B */
        acc = __builtin_amdgcn_wmma_f32_16x16x4_f32(
                  false, a, false, a, (short)0, acc, false, false);
    }

    /* exchange the Gram diagonal through LDS (64B per wave) */
    __shared__ float diag[8][16];
#pragma unroll
    for (int v = 0; v < 8; ++v) {
        const int m = v + 8 * hi;
        if (m16 == m) diag[wave][m] = acc[v];
    }
    __syncthreads();

    const float inv_n = 1.0f / fmaxf(sqrtf(diag[wave][m16]), EPS);
    float sum = 0.f;
#pragma unroll
    for (int v = 0; v < 8; ++v) {
        const int m = v + 8 * hi;       /* row index of this C element  */
        const int n = m16;              /* col index of this C element  */
        if (n > m && n < M_) {
            const float inv_m = 1.0f / fmaxf(sqrtf(diag[wave][m]), EPS);
            sum += acc[v] * inv_m * inv_n;
        }
    }
    for (int off = 16; off > 0; off >>= 1)
        sum += __shfl_xor(sum, off, 32);
    if (lane == 0) ws[WS3_OFF + tile] = sum;
}

// ---------------------------------------------------------------------------
// Kernel 4: single-block deterministic combine of all partials -> 6 scalars.
// ---------------------------------------------------------------------------
__global__ void k4_final(const float* __restrict__ ws, float* __restrict__ out) {
    const int tid = threadIdx.x;  /* 256 threads */
    float sq = 0.f, bce = 0.f, phys = 0.f, rank = 0.f, cs = 0.f, dv = 0.f;
    for (int i = tid; i < K1_BLOCKS; i += 256) {
        const float* p = ws + WS1_OFF + i * 4;
        sq += p[0]; bce += p[1]; phys += p[2]; rank += p[3];
    }
    for (int i = tid; i < K2_BLOCKS; i += 256) cs += ws[WS2_OFF + i];
    for (int i = tid; i < NROWS; i += 256)     dv += ws[WS3_OFF + i];

    __shared__ float sm[6][256];
    sm[0][tid] = sq; sm[1][tid] = bce; sm[2][tid] = phys;
    sm[3][tid] = rank; sm[4][tid] = cs; sm[5][tid] = dv;
    __syncthreads();
    for (int off = 128; off > 0; off >>= 1) {
        if (tid < off)
            for (int k = 0; k < 6; ++k) sm[k][tid] += sm[k][tid + off];
        __syncthreads();
    }
    if (tid == 0) {
        const float invN = 1.0f / (float)NELEM;
        const float invP = 1.0f / ((float)NROWS * (float)NPAIRS);
        const float rec_loss  = sm[0][0] * invN + sm[1][0] * invN;
        const float rank_loss = sm[3][0] * invP;
        const float div_loss  = sm[5][0] * invP;
        const float exp_loss  = 1.0f - sm[4][0] * invN;
        const float phys_loss = sm[2][0] * invN;
        const float total = 1.0f * rec_loss + 0.3f * rank_loss + 0.2f * div_loss
                          + 0.1f * exp_loss + 0.1f * phys_loss;
        out[0] = total;    out[1] = rec_loss; out[2] = rank_loss;
        out[3] = div_loss; out[4] = exp_loss; out[5] = phys_loss;
    }
}

extern "C" void kernel_launch(void* const* d_in, const int* in_sizes, int n_in,
                              void* d_out, int out_size, void* d_ws, size_t ws_size,
                              hipStream_t stream) {
    const float* rec_scores = (const float*)d_in[0];
    const float* tgt_scores = (const float*)d_in[1];
    const float* feas       = (const float*)d_in[2];
    const float* tgt_feas   = (const float*)d_in[3];
    const float* rank_s     = (const float*)d_in[4];
    const float* tgt_rank   = (const float*)d_in[5];
    const float* recs       = (const float*)d_in[6];
    const float* just       = (const float*)d_in[7];
    const float* tgt_expl   = (const float*)d_in[8];
    float* ws  = (float*)d_ws;
    float* out = (float*)d_out;

    k1_scores<<<K1_BLOCKS, 256, 0, stream>>>(rec_scores, tgt_scores, feas,
                                             tgt_feas, rank_s, tgt_rank, ws);
    k2_expl<<<K2_BLOCKS, 256, 0, stream>>>(just, tgt_expl, ws);
    k3_div<<<K3_BLOCKS, 256, 0, stream>>>(recs, ws);
    k4_final<<<1, 256, 0, stream>>>(ws, out);
}